// RegionProposer_19954418057734
// MI455X (gfx1250) — compile-verified
//
#include <hip/hip_runtime.h>
#include <hip/hip_bf16.h>

#define N_NODES   50000              // == 3125 * 16  (M multiple of 16)
#define N_EDGES   800000
#define N_EDGES_SL (N_EDGES + N_NODES)
#define HIDDEN    64
#define HEADS     4
#define HC        (HEADS * HIDDEN)   // 256
#define K1PAD     28                 // 27 padded to multiple of 4
#define NEG_SLOPE 0.2f

typedef float v2f __attribute__((ext_vector_type(2)));
typedef float v8f __attribute__((ext_vector_type(8)));

// ---------- monotone float<->u32 key for atomicMax over floats (incl. negatives)
__device__ __forceinline__ unsigned ordKey(float f) {
    unsigned u = __float_as_uint(f);
    return (u & 0x80000000u) ? ~u : (u | 0x80000000u);
}
__device__ __forceinline__ float ordUnkey(unsigned k) {
    unsigned u = (k & 0x80000000u) ? (k & 0x7FFFFFFFu) : ~k;
    return __uint_as_float(u);
}
#define ORDKEY_NEG_INF 0x007FFFFFu   // ordKey(-inf)

__device__ __forceinline__ float lrelu(float x) { return x > 0.f ? x : NEG_SLOPE * x; }

// ---------------------------------------------------------------------------
// WMMA fp32 GEMM: C[M,NN] = A[M,K] @ B[K,NN], row-major.
//  * B panel staged once per block into LDS, pair-interleaved so each lane's
//    B fragment is ONE ds_load_b64:  sB[(r/2)*NN*2 + c*2 + (r&1)] = B[r*NN+c]
//  * 256-thread blocks = 8 waves; each wave owns a 16-row x (16*NSTRIP)-col
//    strip; A fragment (one global_load_b64/lane) reused across NSTRIP WMMAs.
//  * Software-pipelined: step s+1 loads issue before step s WMMAs fire.
// Preconditions: M%16==0, K%4==0, NN%(16*NSTRIP)==0, K*NN % 1024 == 0.
// MODE 0: plain store.  MODE 1: relu(c + biasRow[col]) (scorer fusion).
// ---------------------------------------------------------------------------
template <int K, int NN, int NSTRIP, int MODE>
__global__ __launch_bounds__(256) void gemm_wmma_f32(
        const float* __restrict__ A, const float* __restrict__ B,
        float* __restrict__ C, int M, const float* __restrict__ biasRow)
{
    constexpr int NSTRIPS = NN / (16 * NSTRIP);   // col-strips per M-tile
    constexpr int STEPS   = K / 4;
    __shared__ float sB[K * NN];                  // pair-interleaved B panel

    // ---- cooperative stage of B into LDS (float4 global reads) ----------
    for (int i = threadIdx.x * 4; i < K * NN; i += 256 * 4) {
        const float4 v = *(const float4*)(B + i);
        const int r = i / NN;
        const int c = i - r * NN;                 // multiple of 4
        float* d = sB + (r >> 1) * (NN * 2) + c * 2 + (r & 1);
        d[0] = v.x; d[2] = v.y; d[4] = v.z; d[6] = v.w;
    }
    __syncthreads();

    // ---- tile assignment (uniform per wave) ------------------------------
    const int lane  = threadIdx.x & 31;
    const int wave  = blockIdx.x * 8 + (threadIdx.x >> 5);
    const int mtile = wave / NSTRIPS;
    const int strip = wave - mtile * NSTRIPS;
    if (mtile >= (M >> 4)) return;                // after barrier: safe

    const int mbase = mtile << 4;
    const int half  = lane >> 4;                  // 0|1 -> K-pair select
    const int l16   = lane & 15;
    const float* __restrict__ aRow = A + (long long)(mbase + l16) * K + 2 * half;
    const int bcol = strip * (16 * NSTRIP) + l16;

    // per-strip LDS base for this lane's column (pair-interleaved)
    const float* __restrict__ sBl = sB + half * (NN * 2) + bcol * 2;

    v8f acc[NSTRIP];
    {
        v8f z = {};
#pragma unroll
        for (int j = 0; j < NSTRIP; ++j) acc[j] = z;
    }

    // ---- software-pipelined K loop --------------------------------------
    v2f a_cur = *(const v2f*)(aRow);              // k = 0
    v2f b_cur[NSTRIP];
#pragma unroll
    for (int j = 0; j < NSTRIP; ++j) b_cur[j] = *(const v2f*)(sBl + j * 32);

#pragma unroll
    for (int s = 0; s < STEPS; ++s) {
        v2f a_nxt;
        v2f b_nxt[NSTRIP];
        if (s + 1 < STEPS) {                      // compile-time per unrolled iter
            a_nxt = *(const v2f*)(aRow + 4 * (s + 1));
            const float* __restrict__ sBn = sBl + (s + 1) * (2 * NN * 2);
#pragma unroll
            for (int j = 0; j < NSTRIP; ++j) b_nxt[j] = *(const v2f*)(sBn + j * 32);
        }
#pragma unroll
        for (int j = 0; j < NSTRIP; ++j) {
            acc[j] = __builtin_amdgcn_wmma_f32_16x16x4_f32(
                /*neg_a=*/false, a_cur, /*neg_b=*/false, b_cur[j],
                /*c_mod=*/(short)0, acc[j], /*reuse_a=*/false, /*reuse_b=*/false);
        }
        a_cur = a_nxt;
#pragma unroll
        for (int j = 0; j < NSTRIP; ++j) b_cur[j] = b_nxt[j];
    }

    // ---- epilogue --------------------------------------------------------
#pragma unroll
    for (int j = 0; j < NSTRIP; ++j) {
        const int col = bcol + 16 * j;
        const float bb = (MODE == 1) ? biasRow[col] : 0.f;
#pragma unroll
        for (int i = 0; i < 8; ++i) {
            const int r = mbase + i + half * 8;
            float v = acc[j][i];
            if (MODE == 1) { v += bb; v = v > 0.f ? v : 0.f; }
            C[(long long)r * NN + col] = v;
        }
    }
}

// ---------------------------------------------------------------------------
// Zero-padding staging: x[50000,27] -> Xp[50000,28], W1[27,256] -> W1p[28,256]
// ---------------------------------------------------------------------------
__global__ void pad_x_k(const float* __restrict__ X, float* __restrict__ Xp)
{
    const int t = blockIdx.x * blockDim.x + threadIdx.x;
    if (t >= N_NODES * K1PAD) return;
    const int n = t / K1PAD, c = t - n * K1PAD;
    Xp[t] = (c < 27) ? X[n * 27 + c] : 0.f;
}

__global__ void pad_w1_k(const float* __restrict__ W1, float* __restrict__ W1p)
{
    const int t = blockIdx.x * blockDim.x + threadIdx.x;
    if (t >= K1PAD * HC) return;
    W1p[t] = (t < 27 * HC) ? W1[t] : 0.f;
}

// ---------------------------------------------------------------------------
// Per (node, head): alpha dot products + softmax-state init (float4 loads).
// ---------------------------------------------------------------------------
__global__ void alpha_init_k(const float* __restrict__ H,
                             const float* __restrict__ a_src, const float* __restrict__ a_dst,
                             float* __restrict__ as, float* __restrict__ ad,
                             unsigned* __restrict__ emax, float* __restrict__ denom)
{
    const int t = blockIdx.x * blockDim.x + threadIdx.x;
    if (t >= N_NODES * HEADS) return;
    const int n = t >> 2, h = t & 3;
    const float* __restrict__ hp  = H + (long long)n * HC + h * HIDDEN;
    const float* __restrict__ asw = a_src + h * HIDDEN;
    const float* __restrict__ adw = a_dst + h * HIDDEN;
    float ss = 0.f, sd = 0.f;
#pragma unroll
    for (int c = 0; c < HIDDEN; c += 4) {
        const float4 v = *(const float4*)(hp + c);
        const float4 s = *(const float4*)(asw + c);
        const float4 d = *(const float4*)(adw + c);
        ss += v.x * s.x + v.y * s.y + v.z * s.z + v.w * s.w;
        sd += v.x * d.x + v.y * d.y + v.z * d.z + v.w * d.w;
    }
    as[t] = ss;
    ad[t] = sd;
    emax[t]  = ORDKEY_NEG_INF;
    denom[t] = 0.f;
}

__global__ void zero_f32v4_k(float* __restrict__ p, int n4)   // n4 = count/4
{
    const int t = blockIdx.x * blockDim.x + threadIdx.x;
    if (t < n4) { float4 z = {0.f, 0.f, 0.f, 0.f}; *(float4*)(p + 4 * t) = z; }
}

__device__ __forceinline__ void edge_src_dst(const int* __restrict__ ei, int e, int& src, int& dst)
{
    if (e < N_EDGES) { src = ei[e]; dst = ei[N_EDGES + e]; }
    else             { src = dst = e - N_EDGES; }   // self loop
}

// Pass 1: segment max via ordered-bit atomicMax.
__global__ void edge_max_k(const int* __restrict__ ei,
                           const float* __restrict__ as, const float* __restrict__ ad,
                           unsigned* __restrict__ emax)
{
    const int t = blockIdx.x * blockDim.x + threadIdx.x;
    if (t >= N_EDGES_SL * HEADS) return;
    const int e = t >> 2, h = t & 3;
    int src, dst; edge_src_dst(ei, e, src, dst);
    const float x = lrelu(as[src * HEADS + h] + ad[dst * HEADS + h]);
    atomicMax(&emax[dst * HEADS + h], ordKey(x));
}

// Pass 2: e_exp per edge + segment denominator.
__global__ void edge_exp_k(const int* __restrict__ ei,
                           const float* __restrict__ as, const float* __restrict__ ad,
                           const unsigned* __restrict__ emax,
                           float* __restrict__ eexp, float* __restrict__ denom)
{
    const int t = blockIdx.x * blockDim.x + threadIdx.x;
    if (t >= N_EDGES_SL * HEADS) return;
    const int e = t >> 2, h = t & 3;
    int src, dst; edge_src_dst(ei, e, src, dst);
    const float x  = lrelu(as[src * HEADS + h] + ad[dst * HEADS + h]);
    const float ex = expf(x - ordUnkey(emax[dst * HEADS + h]));
    eexp[t] = ex;
    atomicAdd(&denom[dst * HEADS + h], ex);
}

// Pass 3: aggregate messages; head-mean (x0.25) folded in -> Y is [N, HIDDEN].
__global__ void edge_agg_k(const int* __restrict__ ei, const float* __restrict__ H,
                           const float* __restrict__ eexp, const float* __restrict__ denom,
                           float* __restrict__ Y)
{
    const long long t = (long long)blockIdx.x * blockDim.x + threadIdx.x;
    if (t >= (long long)N_EDGES_SL * HIDDEN) return;
    const int e = (int)(t >> 6);
    const int c = (int)(t & 63);
    int src, dst; edge_src_dst(ei, e, src, dst);

    const float4 ex = *(const float4*)(eexp  + (size_t)e   * HEADS);  // 16B aligned
    const float4 dn = *(const float4*)(denom + (size_t)dst * HEADS);  // 16B aligned
    const float* __restrict__ hp = H + (long long)src * HC + c;
    __builtin_prefetch(hp, 0, 0);                                     // global_prefetch_b8

    float contrib = (ex.x / dn.x) * hp[0 * HIDDEN]
                  + (ex.y / dn.y) * hp[1 * HIDDEN]
                  + (ex.z / dn.z) * hp[2 * HIDDEN]
                  + (ex.w / dn.w) * hp[3 * HIDDEN];
    atomicAdd(&Y[dst * HIDDEN + c], 0.25f * contrib);
}

// Layer epilogue: elu(Y + bias), float4-vectorized (in-place safe).
__global__ void finalize_elu_k(const float* __restrict__ Y, const float* __restrict__ bias,
                               float* __restrict__ X)
{
    const int t = blockIdx.x * blockDim.x + threadIdx.x;
    if (t >= N_NODES * HIDDEN / 4) return;
    const int idx = t * 4;
    const float4 y = *(const float4*)(Y + idx);
    const float4 b = *(const float4*)(bias + (idx & 63));
    float4 r;
    float v;
    v = y.x + b.x; r.x = v > 0.f ? v : (expf(v) - 1.f);
    v = y.y + b.y; r.y = v > 0.f ? v : (expf(v) - 1.f);
    v = y.z + b.z; r.z = v > 0.f ? v : (expf(v) - 1.f);
    v = y.w + b.w; r.w = v > 0.f ? v : (expf(v) - 1.f);
    *(float4*)(X + idx) = r;
}

// donor_row[c] = bs1[c] + sum_k donor[k] * Ws1[64+k, c]   (folds concat away)
__global__ void donor_row_k(const float* __restrict__ donor, const float* __restrict__ Ws1,
                            const float* __restrict__ bs1, float* __restrict__ dr)
{
    const int c = threadIdx.x;
    if (c >= HIDDEN) return;
    float s = bs1[c];
#pragma unroll
    for (int k = 0; k < 32; ++k) s += donor[k] * Ws1[(HIDDEN + k) * HIDDEN + c];
    dr[c] = s;
}

__global__ void logits_k(const float* __restrict__ hidden, const float* __restrict__ Ws2,
                         const float* __restrict__ bs2, float* __restrict__ out)
{
    const int n = blockIdx.x * blockDim.x + threadIdx.x;
    if (n >= N_NODES) return;
    const float* __restrict__ hp = hidden + (long long)n * HIDDEN;
    float s = bs2[0];
#pragma unroll
    for (int c = 0; c < HIDDEN; c += 4) {
        const float4 h4 = *(const float4*)(hp + c);
        const float4 w4 = *(const float4*)(Ws2 + c);
        s += h4.x * w4.x + h4.y * w4.y + h4.z * w4.z + h4.w * w4.w;
    }
    out[n] = s;
}

// ---------------------------------------------------------------------------
extern "C" void kernel_launch(void* const* d_in, const int* in_sizes, int n_in,
                              void* d_out, int out_size, void* d_ws, size_t ws_size,
                              hipStream_t stream)
{
    const float* x      = (const float*)d_in[0];
    const int*   ei     = (const int*)  d_in[1];
    const float* donor  = (const float*)d_in[2];
    const float* W1     = (const float*)d_in[3];
    const float* a_src1 = (const float*)d_in[4];
    const float* a_dst1 = (const float*)d_in[5];
    const float* b1     = (const float*)d_in[6];
    const float* W2     = (const float*)d_in[7];
    const float* a_src2 = (const float*)d_in[8];
    const float* a_dst2 = (const float*)d_in[9];
    const float* b2     = (const float*)d_in[10];
    const float* Ws1    = (const float*)d_in[11];
    const float* bs1    = (const float*)d_in[12];
    const float* Ws2    = (const float*)d_in[13];
    const float* bs2    = (const float*)d_in[14];
    float* out = (float*)d_out;

    // ---- workspace carve-up (floats; every offset stays 16B-aligned) -----
    float* ws = (float*)d_ws;
    size_t off = 0;
    float*    H     = ws + off; off += (size_t)N_NODES * HC;          // 12.8M
    float*    as    = ws + off; off += (size_t)N_NODES * HEADS;
    float*    ad    = ws + off; off += (size_t)N_NODES * HEADS;
    unsigned* emax  = (unsigned*)(ws + off); off += (size_t)N_NODES * HEADS;
    float*    denom = ws + off; off += (size_t)N_NODES * HEADS;
    float*    eexp  = ws + off; off += (size_t)N_EDGES_SL * HEADS;    // 3.4M
    float*    Ya    = ws + off; off += (size_t)N_NODES * HIDDEN;      // 3.2M
    float*    Yb    = ws + off; off += (size_t)N_NODES * HIDDEN;      // 3.2M
    float*    Xp    = ws + off; off += (size_t)N_NODES * K1PAD;       // 1.4M
    float*    W1p   = ws + off; off += (size_t)K1PAD * HC;
    float*    dr    = ws + off; off += 64;
    float*    hidden = H;   // H is dead after layer-2 aggregation

    const int TPB = 256;
    const int nhBlocks  = (N_NODES * HEADS + TPB - 1) / TPB;
    const int ehBlocks  = (N_EDGES_SL * HEADS + TPB - 1) / TPB;
    const long long aggTotal = (long long)N_EDGES_SL * HIDDEN;
    const int aggBlocks = (int)((aggTotal + TPB - 1) / TPB);
    const int ncBlocks4 = (N_NODES * HIDDEN / 4 + TPB - 1) / TPB;

    // GEMM geometry: 8 waves / 256-thr block; wave strip = 16 x 64
    const int mtiles = N_NODES / 16;                               // 3125
    const int gemmBlocks256 = (mtiles * (HC / 64) + 7) / 8;        // 1563
    const int gemmBlocks64  = (mtiles * (HIDDEN / 64) + 7) / 8;    // 391

    // ------------------------- input staging ------------------------------
    pad_x_k <<<(N_NODES * K1PAD + TPB - 1) / TPB, TPB, 0, stream>>>(x, Xp);
    pad_w1_k<<<(K1PAD * HC + TPB - 1) / TPB, TPB, 0, stream>>>(W1, W1p);

    // =========================== GAT layer 1 ==============================
    gemm_wmma_f32<K1PAD, HC, 4, 0><<<gemmBlocks256, 256, 0, stream>>>(Xp, W1p, H, N_NODES, nullptr);
    alpha_init_k <<<nhBlocks, TPB, 0, stream>>>(H, a_src1, a_dst1, as, ad, emax, denom);
    zero_f32v4_k <<<ncBlocks4, TPB, 0, stream>>>(Ya, N_NODES * HIDDEN / 4);
    edge_max_k   <<<ehBlocks, TPB, 0, stream>>>(ei, as, ad, emax);
    edge_exp_k   <<<ehBlocks, TPB, 0, stream>>>(ei, as, ad, emax, eexp, denom);
    edge_agg_k   <<<aggBlocks, TPB, 0, stream>>>(ei, H, eexp, denom, Ya);
    finalize_elu_k<<<ncBlocks4, TPB, 0, stream>>>(Ya, b1, Ya);

    // =========================== GAT layer 2 ==============================
    gemm_wmma_f32<HIDDEN, HC, 4, 0><<<gemmBlocks256, 256, 0, stream>>>(Ya, W2, H, N_NODES, nullptr);
    alpha_init_k <<<nhBlocks, TPB, 0, stream>>>(H, a_src2, a_dst2, as, ad, emax, denom);
    zero_f32v4_k <<<ncBlocks4, TPB, 0, stream>>>(Yb, N_NODES * HIDDEN / 4);
    edge_max_k   <<<ehBlocks, TPB, 0, stream>>>(ei, as, ad, emax);
    edge_exp_k   <<<ehBlocks, TPB, 0, stream>>>(ei, as, ad, emax, eexp, denom);
    edge_agg_k   <<<aggBlocks, TPB, 0, stream>>>(ei, H, eexp, denom, Yb);
    finalize_elu_k<<<ncBlocks4, TPB, 0, stream>>>(Yb, b2, Yb);

    // ============================== scorer ================================
    donor_row_k  <<<1, 64, 0, stream>>>(donor, Ws1, bs1, dr);
    gemm_wmma_f32<HIDDEN, HIDDEN, 4, 1><<<gemmBlocks64, 256, 0, stream>>>(Yb, Ws1, hidden, N_NODES, dr);
    logits_k     <<<(N_NODES + TPB - 1) / TPB, TPB, 0, stream>>>(hidden, Ws2, bs2, out);
}